// SedMLP_55886114455797
// MI455X (gfx1250) — compile-verified
//
#include <hip/hip_runtime.h>

#define LAMBD 0.01f
#define EDIM 512
#define NBS  8192   // B*S = 4*2048

typedef float v2f  __attribute__((ext_vector_type(2)));
typedef float v4f  __attribute__((ext_vector_type(4)));
typedef float v8f  __attribute__((ext_vector_type(8)));

// Cayley-Dickson tables (compile-time constants; K[i][j] == i^j)
static constexpr int Kt[8][8] = {
    {0,1,2,3,4,5,6,7},{1,0,3,2,5,4,7,6},{2,3,0,1,6,7,4,5},{3,2,1,0,7,6,5,4},
    {4,5,6,7,0,1,2,3},{5,4,7,6,1,0,3,2},{6,7,4,5,2,3,0,1},{7,6,5,4,3,2,1,0}};
static constexpr int SINt[8][8] = {
    { 1,-1,-1,-1,-1,-1,-1,-1},
    { 1, 1,-1,-1,-1, 1,-1,-1},
    { 1,-1, 1, 1,-1,-1, 1,-1},
    { 1,-1,-1, 1,-1,-1,-1, 1},
    { 1,-1,-1,-1, 1, 1, 1, 1},
    { 1, 1,-1,-1,-1, 1, 1,-1},
    { 1,-1, 1,-1,-1, 1, 1,-1},
    { 1,-1, 1,-1,-1,-1,-1, 1}};
static constexpr int SOUTt[8][8] = {
    { 1,-1,-1,-1,-1,-1,-1,-1},
    { 1, 1, 1,-1,-1, 1, 1,-1},
    { 1,-1, 1, 1,-1, 1, 1,-1},
    { 1, 1,-1, 1,-1, 1,-1, 1},
    { 1,-1,-1,-1, 1, 1, 1, 1},
    { 1, 1,-1, 1,-1, 1, 1,-1},
    { 1, 1, 1,-1,-1, 1, 1,-1},
    { 1,-1, 1,-1,-1, 1,-1, 1}};

// softshrink(x) = x - clamp(x,-l,l)  ->  v_med3_f32 + v_sub_f32 (2 VALU ops)
__device__ __forceinline__ float softshrink(float x) {
    return x - fminf(fmaxf(x, -LAMBD), LAMBD);
}

// ---------------------------------------------------------------------------
// Kernel 1: row sums of an (8,E,E) tensor -> (8,E), via WMMA D = A * ones.
// 4096 rows total; each 32-thread block reduces 16 rows with 128 chained
// V_WMMA_F32_16X16X4_F32 ops (B = all-ones => D[m][*] = row-sum of A block).
// A layout per ISA 7.12.2 (32-bit A 16x4): VGPR0 lanes0-15=K0 /lanes16-31=K2,
// VGPR1 lanes0-15=K1 / lanes16-31=K3  => float2 load at col + (hi?2:0).
// ---------------------------------------------------------------------------
__global__ void __launch_bounds__(32) rowsum_kernel(const float* __restrict__ y,
                                                    float* __restrict__ w)
{
    const int lane    = threadIdx.x;          // 0..31, wave32
    const int rowBase = blockIdx.x * 16;      // 16 rows per wave
#if __has_builtin(__builtin_amdgcn_wmma_f32_16x16x4_f32)
    const int  m  = lane & 15;
    const bool hi = lane >= 16;
    const float* rowPtr = y + (size_t)(rowBase + m) * EDIM + (hi ? 2 : 0);

    v2f ones; ones.x = 1.0f; ones.y = 1.0f;
    v8f acc = {};
    for (int col = 0; col < EDIM; col += 4) {
        float2 d = *(const float2*)(rowPtr + col);
        v2f a; a.x = d.x; a.y = d.y;
        acc = __builtin_amdgcn_wmma_f32_16x16x4_f32(
                  false, a, false, ones, (short)0, acc, false, false);
    }
    // D layout: VGPR r: lanes0-15 -> (M=r, N=lane); lanes16-31 -> (M=r+8).
    // Lane 0 holds rows 0..7 (col 0), lane 16 holds rows 8..15 (col 0).
    if (lane == 0) {
#pragma unroll
        for (int r = 0; r < 8; ++r) w[rowBase + r] = acc[r];
    } else if (lane == 16) {
#pragma unroll
        for (int r = 0; r < 8; ++r) w[rowBase + 8 + r] = acc[r];
    }
#else
    // VALU fallback: lanes 0..15 each serially reduce one row.
    if (lane < 16) {
        const float* p = y + (size_t)(rowBase + lane) * EDIM;
        float s = 0.0f;
        for (int c = 0; c < EDIM; ++c) s += p[c];
        w[rowBase + lane] = s;
    }
#endif
}

// ---------------------------------------------------------------------------
// Kernel 2: fused sedenion elementwise. One thread handles one (b,s) slice
// and 4 consecutive e values: loads 16 float4 of x once, produces all 8
// outputs (real+imag). Weights staged in LDS (32 KB / workgroup).
// ---------------------------------------------------------------------------
__global__ void __launch_bounds__(256) sed_main_kernel(
        const float* __restrict__ xr,   const float* __restrict__ xi,
        const float* __restrict__ wR,   const float* __restrict__ wI,
        const float* __restrict__ biasR,const float* __restrict__ biasI,
        float* __restrict__ out)
{
    __shared__ float s_wR[8 * EDIM];
    __shared__ float s_wI[8 * EDIM];
    for (int t = threadIdx.x; t < (8 * EDIM) / 4; t += 256) {
        ((v4f*)s_wR)[t] = ((const v4f*)wR)[t];
        ((v4f*)s_wI)[t] = ((const v4f*)wI)[t];
    }
    __syncthreads();

    const int gid = blockIdx.x * 256 + threadIdx.x;   // 0 .. NBS*(E/4)-1
    const int e   = (gid & (EDIM / 4 - 1)) * 4;
    const int bs  = gid >> 7;                         // (E/4)==128 per slice

    // Load x once (single-touch stream -> non-temporal).
    v4f xrv[8], xiv[8];
    const float* xrb = xr + (size_t)bs * (8 * EDIM) + e;
    const float* xib = xi + (size_t)bs * (8 * EDIM) + e;
#pragma unroll
    for (int j = 0; j < 8; ++j) {
        xrv[j] = __builtin_nontemporal_load((const v4f*)(xrb + j * EDIM));
        xiv[j] = __builtin_nontemporal_load((const v4f*)(xib + j * EDIM));
    }

#pragma unroll
    for (int i = 0; i < 8; ++i) {
        v4f cr = {0.f, 0.f, 0.f, 0.f};
        v4f ci = {0.f, 0.f, 0.f, 0.f};
#pragma unroll
        for (int j = 0; j < 8; ++j) {
            const int k = Kt[i][j];                   // == i ^ j, folded
            v4f wr  = *(const v4f*)(&s_wR[k * EDIM + e]);
            v4f wi  = *(const v4f*)(&s_wI[k * EDIM + e]);
            v4f wis = (SINt[i][j] > 0) ? wi : -wi;    // sign folds into fma
            v4f a = xrv[j] * wr  - xiv[j] * wis;      // real part pre-shrink
            v4f b = xrv[j] * wis + xiv[j] * wr;       // imag part pre-shrink
#pragma unroll
            for (int c = 0; c < 4; ++c) {
                float r  = softshrink(a[c]);
                float im = softshrink(b[c]);
                if (SOUTt[i][j] > 0) { cr[c] += r; ci[c] += im; }
                else                 { cr[c] -= r; ci[c] -= im; }
            }
        }
        // Faithful to source: real gets biasI[i], imag gets biasR[i].
        v4f bI = *(const v4f*)(biasI + i * EDIM + e);
        v4f bR = *(const v4f*)(biasR + i * EDIM + e);
        v4f o0, o1;
        o0[0] = softshrink(cr[0] + bI[0]); o0[1] = softshrink(ci[0] + bR[0]);
        o0[2] = softshrink(cr[1] + bI[1]); o0[3] = softshrink(ci[1] + bR[1]);
        o1[0] = softshrink(cr[2] + bI[2]); o1[1] = softshrink(ci[2] + bR[2]);
        o1[2] = softshrink(cr[3] + bI[3]); o1[3] = softshrink(ci[3] + bR[3]);
        // out shape (8,B,S,E,2): offset = ((i*NBS + bs)*E + e)*2
        float* ob = out + ((size_t)i * NBS + bs) * (size_t)(EDIM * 2) + (size_t)e * 2;
        __builtin_nontemporal_store(o0, (v4f*)ob);
        __builtin_nontemporal_store(o1, (v4f*)(ob + 4));
    }
}

extern "C" void kernel_launch(void* const* d_in, const int* in_sizes, int n_in,
                              void* d_out, int out_size, void* d_ws, size_t ws_size,
                              hipStream_t stream) {
    const float* xr    = (const float*)d_in[0];   // (B,S,8,E)
    const float* xi    = (const float*)d_in[1];   // (B,S,8,E)
    const float* yr    = (const float*)d_in[2];   // (8,E,E)
    const float* yi    = (const float*)d_in[3];   // (8,E,E)
    const float* biasR = (const float*)d_in[4];   // (8,E)
    const float* biasI = (const float*)d_in[5];   // (8,E)
    float* out = (float*)d_out;                   // (8,B,S,E,2)

    float* wR = (float*)d_ws;                     // (8,E) row sums of yr
    float* wI = wR + 8 * EDIM;                    // (8,E) row sums of yi

    // 4096 rows per tensor / 16 rows per wave = 256 blocks of one wave32.
    rowsum_kernel<<<256, 32, 0, stream>>>(yr, wR);
    rowsum_kernel<<<256, 32, 0, stream>>>(yi, wI);

    // NBS * (E/4) threads = 1,048,576 -> 4096 blocks of 256.
    sed_main_kernel<<<4096, 256, 0, stream>>>(xr, xi, wR, wI, biasR, biasI, out);
}